// GNN_11982958756503
// MI455X (gfx1250) — compile-verified
//
#include <hip/hip_runtime.h>
#include <hip/hip_bf16.h>

#define F_IN   32
#define F_E    16
#define D      64
#define L      3
#define NEG_SLOPE 0.2f
#define BN_EPS 1e-5f

typedef __attribute__((ext_vector_type(2))) float v2f;
typedef __attribute__((ext_vector_type(8))) float v8f;

// ---------- ordered-int encoding for float atomic max ----------
__device__ __forceinline__ int f2ord(float f) {
  int i = __float_as_int(f);
  return i >= 0 ? i : (i ^ 0x7FFFFFFF);
}
__device__ __forceinline__ float ord2f(int i) {
  return __int_as_float(i >= 0 ? i : (i ^ 0x7FFFFFFF));
}

// ---------- self-loop edge_attr mean (layer invariant) ----------
__global__ void k_loop_accum(const float* __restrict__ eattr, const int* __restrict__ dst,
                             float* __restrict__ lsum, float* __restrict__ deg, int E) {
  int e = blockIdx.x * blockDim.x + threadIdx.x;
  if (e >= E) return;
  int d = dst[e];
  #pragma unroll
  for (int f = 0; f < F_E; f++) atomicAdd(&lsum[d * F_E + f], eattr[e * F_E + f]);
  atomicAdd(&deg[d], 1.0f);
}

__global__ void k_loop_div(float* __restrict__ lsum, const float* __restrict__ deg, int n) {
  int i = blockIdx.x * blockDim.x + threadIdx.x;
  if (i >= n * F_E) return;
  float c = deg[i / F_E];
  lsum[i] = lsum[i] / fmaxf(c, 1.0f);
}

// ---------- w_ea[l] = We[l] @ a_e[l]  (L x F_E values) ----------
__global__ void k_wea(const float* __restrict__ We, const float* __restrict__ ae,
                      float* __restrict__ wea) {
  int i = blockIdx.x * blockDim.x + threadIdx.x;
  if (i >= L * F_E) return;
  int l = i / F_E, f = i % F_E;
  float s = 0.f;
  #pragma unroll
  for (int d = 0; d < D; d++) s += We[(l * F_E + f) * D + d] * ae[l * D + d];
  wea[i] = s;
}

// ---------- fp32 WMMA GEMM: C[n x 64] = A[n x K] @ B[K x 64] (+bias) ----------
// One wave handles 16 rows x 64 cols using V_WMMA_F32_16X16X4_F32.
__global__ void k_gemm_wmma(const float* __restrict__ A, const float* __restrict__ B,
                            const float* __restrict__ bias, float* __restrict__ C,
                            int n, int K) {
  int wave = (blockIdx.x * blockDim.x + threadIdx.x) >> 5;
  int lane = threadIdx.x & 31;
  int row0 = wave * 16;
  if (row0 >= n) return;                    // wave-uniform exit, EXEC stays full
  int lm = lane & 15;
  int kh = (lane >> 4) << 1;                // 0 for lanes 0-15, 2 for lanes 16-31
  int mrow = row0 + lm;
  int mrd = mrow < n ? mrow : n - 1;        // clamp (n % 16 == 0 in practice)

  v8f acc[4] = {};
  for (int k = 0; k < K; k += 4) {
    v2f a;
    a.x = A[(size_t)mrd * K + k + kh];
    a.y = A[(size_t)mrd * K + k + kh + 1];
    #pragma unroll
    for (int nt = 0; nt < 4; nt++) {
      int col = nt * 16 + lm;
      v2f b;
      b.x = B[(k + kh) * D + col];
      b.y = B[(k + kh + 1) * D + col];
      acc[nt] = __builtin_amdgcn_wmma_f32_16x16x4_f32(
          false, a, false, b, (short)0, acc[nt], false, false);
    }
  }

  int rbase = row0 + ((lane >> 4) << 3);    // C/D layout: VGPR r -> M = r + 8*(lane/16)
  #pragma unroll
  for (int nt = 0; nt < 4; nt++) {
    int col = nt * 16 + lm;
    float bv = bias ? bias[col] : 0.f;
    #pragma unroll
    for (int r = 0; r < 8; r++) {
      int row = rbase + r;
      if (row < n) C[(size_t)row * D + col] = acc[nt][r] + bv;
    }
  }
}

// ---------- per-node attention scores s = xw.a_s, d = xw.a_d ----------
__global__ void k_scores(const float* __restrict__ xw, const float* __restrict__ as,
                         const float* __restrict__ ad, float* __restrict__ s,
                         float* __restrict__ dv, int n) {
  int i = blockIdx.x * blockDim.x + threadIdx.x;
  if (i >= n) return;
  const float* row = xw + (size_t)i * D;
  float ss = 0.f, dd = 0.f;
  #pragma unroll
  for (int f = 0; f < D; f++) { float v = row[f]; ss += v * as[f]; dd += v * ad[f]; }
  s[i] = ss;
  dv[i] = dd;
}

__global__ void k_init_amax(int* __restrict__ amax, int n) {
  int i = blockIdx.x * blockDim.x + threadIdx.x;
  if (i < n) amax[i] = (int)0x80000000;
}

// ---------- alpha = leaky_relu(a_src + a_dst + a_edge); segment max over dst ----------
__global__ void k_alpha(const float* __restrict__ s, const float* __restrict__ dv,
                        const float* __restrict__ eattr, const float* __restrict__ lattr,
                        const float* __restrict__ wea_l, const int* __restrict__ src,
                        const int* __restrict__ dst, float* __restrict__ alpha,
                        int* __restrict__ amax, int E, int n) {
  int i = blockIdx.x * blockDim.x + threadIdx.x;
  if (i >= E + n) return;
  int se, de;
  const float* ea;
  if (i < E) { se = src[i]; de = dst[i]; ea = eattr + (size_t)i * F_E; }
  else       { se = i - E;  de = i - E;  ea = lattr + (size_t)(i - E) * F_E; }
  float aed = 0.f;
  #pragma unroll
  for (int f = 0; f < F_E; f++) aed += ea[f] * wea_l[f];
  float a = s[se] + dv[de] + aed;
  a = a > 0.f ? a : NEG_SLOPE * a;
  alpha[i] = a;
  atomicMax(&amax[de], f2ord(a));
}

// ---------- ex = exp(alpha - amax[dst]); denom[dst] += ex ----------
__global__ void k_exp(float* __restrict__ alpha, const int* __restrict__ amax,
                      const int* __restrict__ dst, float* __restrict__ denom, int E, int n) {
  int i = blockIdx.x * blockDim.x + threadIdx.x;
  if (i >= E + n) return;
  int de = i < E ? dst[i] : i - E;
  float ex = expf(alpha[i] - ord2f(amax[de]));
  alpha[i] = ex;
  atomicAdd(&denom[de], ex);
}

// ---------- acc[dst] += (ex/denom[dst]) * xw[src] ; 16 threads/edge x 4 feats ----------
__global__ void k_aggr(const float* __restrict__ ex, const float* __restrict__ denom,
                       const float* __restrict__ xw, const int* __restrict__ src,
                       const int* __restrict__ dst, float* __restrict__ acc, int E, int n) {
  int t = blockIdx.x * blockDim.x + threadIdx.x;
  if (t >= (E + n) * 16) return;
  int i = t >> 4;
  int f0 = (t & 15) << 2;
  int se = i < E ? src[i] : i - E;
  int de = i < E ? dst[i] : i - E;
  float w = ex[i] / denom[de];
  const float4 v = *(const float4*)(xw + (size_t)se * D + f0);
  float* ap = acc + (size_t)de * D + f0;
  atomicAdd(ap + 0, w * v.x);
  atomicAdd(ap + 1, w * v.y);
  atomicAdd(ap + 2, w * v.z);
  atomicAdd(ap + 3, w * v.w);
}

// ---------- h = BN(acc + bias) [+ ReLU] ----------
__global__ void k_bn(const float* __restrict__ acc, const float* __restrict__ bias,
                     const float* __restrict__ gamma, const float* __restrict__ beta,
                     const float* __restrict__ mean, const float* __restrict__ var,
                     float* __restrict__ h, int n, int relu) {
  int i = blockIdx.x * blockDim.x + threadIdx.x;
  if (i >= n * D) return;
  int f = i & (D - 1);
  float v = acc[i] + bias[f];
  v = (v - mean[f]) * (gamma[f] / sqrtf(var[f] + BN_EPS)) + beta[f];
  if (relu) v = fmaxf(v, 0.f);
  h[i] = v;
}

// ---------- node output copy + atomic graph readout ----------
__global__ void k_output(const float* __restrict__ h, const int* __restrict__ batch,
                         float* __restrict__ out_nodes, float* __restrict__ out_graph, int n) {
  int i = blockIdx.x * blockDim.x + threadIdx.x;
  if (i >= n * D) return;
  float v = h[i];
  out_nodes[i] = v;
  int node = i >> 6;
  int f = i & (D - 1);
  atomicAdd(&out_graph[(size_t)batch[node] * D + f], v);
}

extern "C" void kernel_launch(void* const* d_in, const int* in_sizes, int n_in,
                              void* d_out, int out_size, void* d_ws, size_t ws_size,
                              hipStream_t stream) {
  const float* x     = (const float*)d_in[0];
  const int*   eidx  = (const int*)d_in[1];
  const float* eattr = (const float*)d_in[2];
  const int*   batch = (const int*)d_in[3];
  const float* xlw   = (const float*)d_in[4];
  const float* xlb   = (const float*)d_in[5];
  const float* glw   = (const float*)d_in[6];
  const float* gas   = (const float*)d_in[7];
  const float* gad   = (const float*)d_in[8];
  const float* gew   = (const float*)d_in[9];
  const float* gae   = (const float*)d_in[10];
  const float* gb    = (const float*)d_in[11];
  const float* bng   = (const float*)d_in[12];
  const float* bnb   = (const float*)d_in[13];
  const float* bnm   = (const float*)d_in[14];
  const float* bnv   = (const float*)d_in[15];

  const int n = in_sizes[3];          // N nodes (batch length)
  const int E = in_sizes[2] / F_E;    // edges
  const int* srcp = eidx;
  const int* dstp = eidx + E;

  // workspace layout (floats)
  float* ws    = (float*)d_ws;
  float* h     = ws;  ws += (size_t)n * D;
  float* xw    = ws;  ws += (size_t)n * D;
  float* acc   = ws;  ws += (size_t)n * D;
  float* sS    = ws;  ws += n;
  float* dS    = ws;  ws += n;
  float* lattr = ws;  ws += (size_t)n * F_E;
  float* deg   = ws;  ws += n;
  int*   amax  = (int*)ws; ws += n;
  float* denom = ws;  ws += n;
  float* alpha = ws;  ws += (size_t)(E + n);
  float* wea   = ws;  ws += L * F_E;

  float* out_nodes = (float*)d_out;
  float* out_graph = out_nodes + (size_t)n * D;
  const size_t graph_elems = (size_t)out_size - (size_t)n * D;

  const int B = 256;
  #define GRID(x) (((x) + B - 1) / B)

  // --- layer-invariant precompute ---
  hipMemsetAsync(lattr, 0, (size_t)n * F_E * sizeof(float), stream);
  hipMemsetAsync(deg, 0, (size_t)n * sizeof(float), stream);
  k_loop_accum<<<GRID(E), B, 0, stream>>>(eattr, dstp, lattr, deg, E);
  k_loop_div<<<GRID(n * F_E), B, 0, stream>>>(lattr, deg, n);
  k_wea<<<GRID(L * F_E), B, 0, stream>>>(gew, gae, wea);

  // --- input linear: h = x @ xlw + xlb (WMMA fp32) ---
  {
    int waves = (n + 15) / 16;
    int blocks = (waves + 3) / 4;   // 4 waves / 128-thread block
    k_gemm_wmma<<<blocks, 128, 0, stream>>>(x, xlw, xlb, h, n, F_IN);
  }

  // --- GAT layers ---
  for (int l = 0; l < L; l++) {
    hipMemsetAsync(denom, 0, (size_t)n * sizeof(float), stream);
    hipMemsetAsync(acc, 0, (size_t)n * D * sizeof(float), stream);
    k_init_amax<<<GRID(n), B, 0, stream>>>(amax, n);

    int waves = (n + 15) / 16;
    int blocks = (waves + 3) / 4;
    k_gemm_wmma<<<blocks, 128, 0, stream>>>(h, glw + (size_t)l * D * D, nullptr, xw, n, D);

    k_scores<<<GRID(n), B, 0, stream>>>(xw, gas + (size_t)l * D, gad + (size_t)l * D, sS, dS, n);
    k_alpha<<<GRID(E + n), B, 0, stream>>>(sS, dS, eattr, lattr, wea + (size_t)l * F_E,
                                           srcp, dstp, alpha, amax, E, n);
    k_exp<<<GRID(E + n), B, 0, stream>>>(alpha, amax, dstp, denom, E, n);
    k_aggr<<<GRID((E + n) * 16), B, 0, stream>>>(alpha, denom, xw, srcp, dstp, acc, E, n);
    k_bn<<<GRID(n * D), B, 0, stream>>>(acc, gb + (size_t)l * D, bng + (size_t)l * D,
                                        bnb + (size_t)l * D, bnm + (size_t)l * D,
                                        bnv + (size_t)l * D, h, n, l < L - 1 ? 1 : 0);
  }

  // --- outputs ---
  hipMemsetAsync(out_graph, 0, graph_elems * sizeof(float), stream);
  k_output<<<GRID(n * D), B, 0, stream>>>(h, batch, out_nodes, out_graph, n);
  #undef GRID
}